// CenterHeadTemplate_8753143349332
// MI455X (gfx1250) — compile-verified
//
#include <hip/hip_runtime.h>
#include <cstdint>

// ---------------------------------------------------------------------------
// CenterPoint decode head for MI455X (gfx1250).
//
// Roofline: inputs ~151MB; heat (84MB) is read twice, but L2 is 192MB so the
// second pass hits L2 (default RT temporal hint keeps it resident) -> HBM
// traffic ~= 84MB + ~6MB aux ~= 4us at 23.3TB/s. No matmul exists in this
// workload, so WMMA is inapplicable; the CDNA5 paths that matter are:
//   * global_load_async_to_lds_b128 (ASYNCcnt) staging of stencil tiles
//   * s_wait_asynccnt split counter
//   * LDS atomics + LDS-resident bitonic sorts (320KB WGP LDS)
//   * wave32 blocks
// Histogram partials are written with plain stores (no global atomics) and
// reduced in the threshold kernel: avoids ~21M L2 atomic RMWs.
// ---------------------------------------------------------------------------

namespace {
constexpr int BATCH  = 8;
constexpr int NCLS   = 10;
constexpr int HEIGHT = 512;
constexpr int WIDTH  = 512;
constexpr int HWSZ   = HEIGHT * WIDTH;
constexpr int KPROP  = 500;
constexpr int NPLANE = BATCH * NCLS;        // 80 heat planes
constexpr int NBINS  = 1024;                // score histogram bins over (0,1)
constexpr int CAP    = 4096;                // candidate buffer per plane
constexpr int SROWS  = 32;                  // strip height (full 512 width)
constexpr int NSTRIP = HEIGHT / SROWS;      // 16 strips per plane
constexpr int TROWS  = SROWS + 2;           // 34 rows incl. halo
constexpr int TCOLS  = WIDTH + 8;           // 520: 16B-aligned halo (-4..515)
constexpr int NSEG   = TROWS * (TCOLS / 4); // 4420 16-byte segments per strip
constexpr int NTHR   = 1024;                // 32 waves per strip block

constexpr size_t PART_BYTES = (size_t)NPLANE * NSTRIP * NBINS * 4; // 5,242,880
constexpr size_t THR_BYTES  = (size_t)NPLANE * 4;
constexpr size_t CNT_BYTES  = (size_t)NPLANE * 4;
constexpr size_t PREFIX     = PART_BYTES + THR_BYTES + CNT_BYTES;  // 8B-mult
constexpr size_t CAND_BYTES = (size_t)NPLANE * CAP * 8;
} // namespace

// -------- CDNA5 async global->LDS copy (16 bytes per lane, ASYNCcnt) --------
__device__ __forceinline__ void async_b128_to_lds(uint32_t lds_byte_addr,
                                                  const void* gptr) {
  unsigned long long ga = (unsigned long long)(uintptr_t)gptr;
  asm volatile("global_load_async_to_lds_b128 %0, %1, off"
               :: "v"(lds_byte_addr), "v"(ga) : "memory");
}
__device__ __forceinline__ void wait_async0() {
  asm volatile("s_wait_asynccnt 0" ::: "memory");
}

// ---------------------------------------------------------------------------
// Pass 1 / Pass 3: sigmoid + 3x3 NMS on one 512x32 strip of one (b,c) plane.
// PASS1: private LDS histogram -> plain-store per-strip partial to global.
// !PASS1: compact (score, index) keys above per-plane threshold bin.
// Key packing: [63:32]=float bits of score (score in (0,1) -> monotonic),
//              [31:0]=~index -> descending u64 sort == score desc, index asc
//              (matches lax.top_k lowest-index tie-break).
// ---------------------------------------------------------------------------
template <bool PASS1>
__global__ void nms_pass(const float* __restrict__ heat,
                         unsigned* __restrict__ partHist,
                         const int* __restrict__ thrBin,
                         unsigned* __restrict__ candCount,
                         unsigned long long* __restrict__ cand) {
  const int strip   = blockIdx.x;
  const int p       = blockIdx.y;                 // plane = b*NCLS + c
  const int rowBase = strip * SROWS;
  const float* plane = heat + (size_t)p * HWSZ;
  const int tid = threadIdx.x;

  __shared__ __align__(16) float tile[TROWS * TCOLS];  // ~69KB of 320KB WGP LDS
  __shared__ unsigned shist[NBINS];

  if (PASS1) {
    for (int i = tid; i < NBINS; i += NTHR) shist[i] = 0u;
  }

  // ---- stage halo strip into LDS: interior 16B segments via async engine ----
  for (int seg = tid; seg < NSEG; seg += NTHR) {
    const int r   = seg / (TCOLS / 4);
    const int s   = seg % (TCOLS / 4);
    const int gr  = rowBase - 1 + r;
    const int gc0 = s * 4 - 4;
    float* dst = &tile[r * TCOLS + s * 4];
    if (gr >= 0 && gr < HEIGHT && gc0 >= 0 && gc0 + 3 < WIDTH) {
      // RT temporal hint on purpose: keeps heat resident in 192MB L2 so the
      // second NMS pass is served from L2 instead of HBM.
      async_b128_to_lds((uint32_t)(uintptr_t)dst,
                        plane + (size_t)gr * WIDTH + gc0);
    } else {
#pragma unroll
      for (int e = 0; e < 4; ++e) {
        const int gc = gc0 + e;
        dst[e] = (gr >= 0 && gr < HEIGHT && gc >= 0 && gc < WIDTH)
                     ? plane[(size_t)gr * WIDTH + gc]
                     : -__builtin_inff();         // -inf pad == reduce_window
      }
    }
  }
  wait_async0();
  __syncthreads();

  const int thr = PASS1 ? 0 : thrBin[p];

  // ---- 3x3 max stencil from LDS; sigmoid only on kept peaks ----
  for (int px = tid; px < SROWS * WIDTH; px += NTHR) {
    const int y = px >> 9, x = px & (WIDTH - 1);
    const float* t0 = &tile[y * TCOLS + (x + 3)];  // halo row above, col lc-1
    const float* t1 = t0 + TCOLS;
    const float* t2 = t1 + TCOLS;
    const float raw = t1[1];
    float m = fmaxf(fmaxf(t0[0], t0[1]), t0[2]);
    m = fmaxf(m, fmaxf(fmaxf(t1[0], raw), t1[2]));
    m = fmaxf(m, fmaxf(fmaxf(t2[0], t2[1]), t2[2]));
    // sigmoid monotone-injective: (sigmoid(raw)==maxpool(sigmoid)) <=> raw==m
    const float score = (raw == m) ? 1.0f / (1.0f + __expf(-raw)) : 0.0f;
    int bin = (int)(score * (float)NBINS);
    bin = bin > (NBINS - 1) ? (NBINS - 1) : (bin < 0 ? 0 : bin);
    if (PASS1) {
      atomicAdd(&shist[bin], 1u);                  // ds_add, no global atomics
    } else if (score > 0.0f && bin >= thr) {
      const unsigned slot = atomicAdd(&candCount[p], 1u); // ~600/plane total
      if (slot < CAP) {
        const unsigned idx = (unsigned)((rowBase + y) * WIDTH + x);
        cand[(size_t)p * CAP + slot] =
            ((unsigned long long)__float_as_uint(score) << 32) |
            (unsigned long long)(0xFFFFFFFFu - idx);
      }
    }
  }

  if (PASS1) {
    __syncthreads();
    unsigned* dstHist = partHist + ((size_t)p * NSTRIP + strip) * NBINS;
    for (int i = tid; i < NBINS; i += NTHR) dstHist[i] = shist[i]; // plain store
  }
}

// ---------------------------------------------------------------------------
// Per-plane threshold: reduce the 16 strip partials, then find largest bin
// whose suffix count reaches KPROP. Bin 0 (suppressed zeros) never accepted.
// ---------------------------------------------------------------------------
__global__ void find_threshold(const unsigned* __restrict__ partHist,
                               int* __restrict__ thrBin) {
  const int p = blockIdx.x;
  __shared__ unsigned comb[NBINS];
  for (int b = threadIdx.x; b < NBINS; b += blockDim.x) {
    unsigned v = 0;
#pragma unroll
    for (int s = 0; s < NSTRIP; ++s)
      v += partHist[((size_t)p * NSTRIP + s) * NBINS + b];
    comb[b] = v;
  }
  __syncthreads();
  if (threadIdx.x == 0) {
    unsigned cum = 0;
    int thr = 1;
    for (int b = NBINS - 1; b >= 1; --b) {
      cum += comb[b];
      if (cum >= (unsigned)KPROP) { thr = b; break; }
    }
    thrBin[p] = thr;
  }
}

// ---------------------------------------------------------------------------
// Per-plane: bitonic-sort <=4096 candidate keys in LDS (32KB), keep top-500.
// ---------------------------------------------------------------------------
__global__ void plane_topk(const unsigned long long* __restrict__ cand,
                           const unsigned* __restrict__ candCount,
                           unsigned long long* __restrict__ ptk) {
  const int p = blockIdx.x;
  __shared__ unsigned long long s[CAP];
  unsigned n = candCount[p];
  if (n > CAP) n = CAP;
  for (int i = threadIdx.x; i < CAP; i += blockDim.x)
    s[i] = (i < (int)n) ? cand[(size_t)p * CAP + i] : 0ull;
  __syncthreads();
  for (unsigned k = 2; k <= (unsigned)CAP; k <<= 1) {
    for (unsigned j = k >> 1; j > 0; j >>= 1) {
      for (unsigned i = threadIdx.x; i < (unsigned)CAP; i += blockDim.x) {
        const unsigned ixj = i ^ j;
        if (ixj > i) {
          const unsigned long long a = s[i], b = s[ixj];
          const bool up = ((i & k) == 0);
          if ((a < b) == up) { s[i] = b; s[ixj] = a; }  // descending overall
        }
      }
      __syncthreads();
    }
  }
  for (int i = threadIdx.x; i < KPROP; i += blockDim.x)
    ptk[(size_t)p * KPROP + i] = s[i];
}

// ---------------------------------------------------------------------------
// Per-batch: merge C*K=5000 class candidates (sort 8192 padded keys in 64KB
// LDS), then gather channels-last features + decode boxes.
// Tie-break key low word = ~(c*K+i) to match lax.top_k lowest-index rule.
// ---------------------------------------------------------------------------
__global__ void batch_merge_boxes(const unsigned long long* __restrict__ ptk,
                                  const float* __restrict__ reg,
                                  const float* __restrict__ hei,
                                  const float* __restrict__ dimm,
                                  const float* __restrict__ rot,
                                  float* __restrict__ out) {
  constexpr int N = 8192;
  const int b = blockIdx.x;
  __shared__ unsigned long long s[N];

  for (int e = threadIdx.x; e < N; e += blockDim.x) {
    unsigned long long key2 = 0ull;
    if (e < NCLS * KPROP) {
      const int c = e / KPROP, i = e % KPROP;
      const unsigned long long orig = ptk[((size_t)b * NCLS + c) * KPROP + i];
      key2 = (orig & 0xFFFFFFFF00000000ull) |
             (unsigned long long)(0xFFFFFFFFu - (unsigned)e);
    }
    s[e] = key2;
  }
  __syncthreads();

  for (unsigned k = 2; k <= (unsigned)N; k <<= 1) {
    for (unsigned j = k >> 1; j > 0; j >>= 1) {
      for (unsigned i = threadIdx.x; i < (unsigned)N; i += blockDim.x) {
        const unsigned ixj = i ^ j;
        if (ixj > i) {
          const unsigned long long a = s[i], c = s[ixj];
          const bool up = ((i & k) == 0);
          if ((a < c) == up) { s[i] = c; s[ixj] = a; }
        }
      }
      __syncthreads();
    }
  }

  for (int k = threadIdx.x; k < KPROP; k += blockDim.x) {
    const unsigned long long key2 = s[k];
    const unsigned pos = 0xFFFFFFFFu - (unsigned)(key2 & 0xFFFFFFFFull);
    const int c = (int)(pos / KPROP), i = (int)(pos % KPROP);
    const unsigned long long orig =
        (pos < (unsigned)(NCLS * KPROP))
            ? ptk[((size_t)b * NCLS + c) * KPROP + i] : 0ull;
    unsigned idx = 0xFFFFFFFFu - (unsigned)(orig & 0xFFFFFFFFull);
    if (idx >= (unsigned)HWSZ) idx = 0;             // padded-entry guard
    const float score = __uint_as_float((unsigned)(orig >> 32));
    const float xs = (float)(idx % WIDTH);
    const float ys = (float)(idx / WIDTH);
    const size_t base = (size_t)b * HWSZ + idx;     // channels-last gathers
    const float r0 = reg[base * 2 + 0], r1 = reg[base * 2 + 1];
    const float x = (xs + r0) * 0.2f - 51.2f;       // STRIDE*VOXEL, PC_MIN
    const float y = (ys + r1) * 0.2f - 51.2f;
    const float z = hei[base];
    const float d0 = __expf(dimm[base * 3 + 0]);
    const float d1 = __expf(dimm[base * 3 + 1]);
    const float d2 = __expf(dimm[base * 3 + 2]);
    const float ang = atan2f(rot[base * 2 + 0], rot[base * 2 + 1]);
    float* o = out + ((size_t)b * KPROP + k) * 8;
    o[0] = x; o[1] = y; o[2] = z; o[3] = d0;
    o[4] = d1; o[5] = d2; o[6] = ang; o[7] = score;
  }
}

// ---------------------------------------------------------------------------
extern "C" void kernel_launch(void* const* d_in, const int* in_sizes, int n_in,
                              void* d_out, int out_size, void* d_ws,
                              size_t ws_size, hipStream_t stream) {
  const float* heat = (const float*)d_in[0];  // (8,10,512,512)
  const float* reg  = (const float*)d_in[1];  // (8,512,512,2)
  const float* hei  = (const float*)d_in[2];  // (8,512,512,1)
  const float* dimm = (const float*)d_in[3];  // (8,512,512,3)
  const float* rot  = (const float*)d_in[4];  // (8,512,512,2)
  float* out = (float*)d_out;                 // (8,500,8)

  char* ws = (char*)d_ws;
  unsigned* part = (unsigned*)ws;
  int* thr       = (int*)(ws + PART_BYTES);
  unsigned* cnt  = (unsigned*)(ws + PART_BYTES + THR_BYTES);
  unsigned long long* cand = (unsigned long long*)(ws + PREFIX);
  unsigned long long* ptk  = (unsigned long long*)(ws + PREFIX + CAND_BYTES);

  // Only candidate counters need re-zeroing (histogram partials are fully
  // overwritten each call). hipMemsetAsync is graph-capturable.
  hipMemsetAsync(cnt, 0, CNT_BYTES, stream);

  const dim3 gStrips(NSTRIP, NPLANE);               // 16 x 80 strip blocks
  nms_pass<true><<<gStrips, NTHR, 0, stream>>>(heat, part, thr, cnt, cand);
  find_threshold<<<NPLANE, 256, 0, stream>>>(part, thr);
  nms_pass<false><<<gStrips, NTHR, 0, stream>>>(heat, part, thr, cnt, cand);
  plane_topk<<<NPLANE, 512, 0, stream>>>(cand, cnt, ptk);
  batch_merge_boxes<<<BATCH, 1024, 0, stream>>>(ptk, reg, hei, dimm, rot, out);
}